// GNNA_GCN_50697793962357
// MI455X (gfx1250) — compile-verified
//
#include <hip/hip_runtime.h>

typedef float v2f __attribute__((ext_vector_type(2)));
typedef float v8f __attribute__((ext_vector_type(8)));

#define N_NODES 100000
#define N_EDGES 3200000
#define IN_F    256
#define HID     128
#define OUT_F   40
#define OUT_P   48   // padded to 3 WMMA column tiles

// ---------------------------------------------------------------- zero fill
__global__ void gcn_zero_f32(float* __restrict__ p, long n) {
    long i = (long)blockIdx.x * blockDim.x + threadIdx.x;
    long stride = (long)gridDim.x * blockDim.x;
    for (; i < n; i += stride) p[i] = 0.0f;
}

// ---------------------------------------------------------------- GEMM0
// hlin[N,128] = x[N,256] @ W0[256,128]
// block = 256 threads = 8 waves; wave w -> rows [ (blk*8+w)*16 , +16 ), all 128 cols.
// W0 staged through LDS in 64x128 panels (32KB).
__global__ __launch_bounds__(256) void gcn_gemm0(const float* __restrict__ x,
                                                 const float* __restrict__ W0,
                                                 float* __restrict__ hlin) {
    __shared__ __align__(16) float lw[64 * HID];   // 32 KB panel of W0
    const int lane = threadIdx.x & 31;
    const int wave = threadIdx.x >> 5;
    const int m0   = (blockIdx.x * 8 + wave) * 16;
    const bool active = (m0 < N_NODES);
    const int col = lane & 15;          // N index within tile / A row within strip
    const int kh  = (lane >> 4) * 2;    // K sub-offset per ISA fragment layout
    const long arow = (long)(m0 + col) * IN_F;

    v8f acc[8];
    const v8f vzero = {0.f,0.f,0.f,0.f,0.f,0.f,0.f,0.f};
    #pragma unroll
    for (int t = 0; t < 8; ++t) acc[t] = vzero;

    for (int p = 0; p < IN_F; p += 64) {
        // cooperative stage: 64 rows x 128 cols of W0 (contiguous) as float4
        const float4* gsrc = (const float4*)(W0 + (long)p * HID);
        for (int i = threadIdx.x; i < (64 * HID) / 4; i += 256)
            ((float4*)lw)[i] = gsrc[i];
        __syncthreads();

        if (active) {
            #pragma unroll 4
            for (int lk = 0; lk < 64; lk += 4) {
                // A fragment: 2 consecutive f32 per lane (K = lk+kh, lk+kh+1)
                v2f a = *(const v2f*)(x + arow + p + lk + kh);
                #pragma unroll
                for (int t = 0; t < 8; ++t) {
                    v2f b;
                    b.x = lw[(lk + kh)     * HID + t * 16 + col];
                    b.y = lw[(lk + kh + 1) * HID + t * 16 + col];
                    acc[t] = __builtin_amdgcn_wmma_f32_16x16x4_f32(
                        false, a, false, b, (short)0, acc[t], false, false);
                }
            }
        }
        __syncthreads();
    }

    if (active) {
        const int moff = (lane >> 4) * 8;   // C/D layout: lanes 16-31 hold M=v+8
        #pragma unroll
        for (int t = 0; t < 8; ++t)
            #pragma unroll
            for (int v = 0; v < 8; ++v)
                hlin[(long)(m0 + moff + v) * HID + t * 16 + col] = acc[t][v];
    }
}

// ---------------------------------------------------------------- SpMM0
// agg0[dst] += hlin[src] * w   (32 threads/edge, float4 slices, coalesced row gather)
__global__ void gcn_spmm0(const float* __restrict__ hlin,
                          const float* __restrict__ ew,
                          const int* __restrict__ src,
                          const int* __restrict__ dst,
                          float* __restrict__ agg0) {
    long gid = (long)blockIdx.x * blockDim.x + threadIdx.x;
    if (gid >= (long)N_EDGES * 32) return;
    const int e = (int)(gid >> 5);
    const int f = (int)(gid & 31) * 4;
    const int s = src[e], d = dst[e];
    const float w = ew[e];
    const float4 v = *(const float4*)(hlin + (long)s * HID + f);
    float* o = agg0 + (long)d * HID + f;
    atomicAdd(o + 0, v.x * w);
    atomicAdd(o + 1, v.y * w);
    atomicAdd(o + 2, v.z * w);
    atomicAdd(o + 3, v.w * w);
}

// ---------------------------------------------------------------- GEMM1
// outlin[N,48] = relu(agg0 + b0) @ W1pad[128,48]   (bias+ReLU fused into A load)
__global__ __launch_bounds__(256) void gcn_gemm1(const float* __restrict__ agg0,
                                                 const float* __restrict__ b0,
                                                 const float* __restrict__ W1,
                                                 float* __restrict__ outlin) {
    __shared__ __align__(16) float lw[HID * OUT_P];   // 24 KB, zero-padded W1
    for (int i = threadIdx.x; i < HID * OUT_P; i += 256) {
        int r = i / OUT_P, c = i - r * OUT_P;
        lw[i] = (c < OUT_F) ? W1[r * OUT_F + c] : 0.0f;
    }
    __syncthreads();

    const int lane = threadIdx.x & 31;
    const int wave = threadIdx.x >> 5;
    const int m0   = (blockIdx.x * 8 + wave) * 16;
    if (m0 >= N_NODES) return;
    const int col = lane & 15;
    const int kh  = (lane >> 4) * 2;
    const long arow = (long)(m0 + col) * HID;

    v8f acc[3];
    const v8f vzero = {0.f,0.f,0.f,0.f,0.f,0.f,0.f,0.f};
    #pragma unroll
    for (int t = 0; t < 3; ++t) acc[t] = vzero;

    #pragma unroll 4
    for (int k = 0; k < HID; k += 4) {
        v2f a  = *(const v2f*)(agg0 + arow + k + kh);
        v2f bb = *(const v2f*)(b0 + k + kh);
        a.x = fmaxf(a.x + bb.x, 0.0f);      // fused bias + ReLU
        a.y = fmaxf(a.y + bb.y, 0.0f);
        #pragma unroll
        for (int t = 0; t < 3; ++t) {
            v2f b;
            b.x = lw[(k + kh)     * OUT_P + t * 16 + col];
            b.y = lw[(k + kh + 1) * OUT_P + t * 16 + col];
            acc[t] = __builtin_amdgcn_wmma_f32_16x16x4_f32(
                false, a, false, b, (short)0, acc[t], false, false);
        }
    }

    const int moff = (lane >> 4) * 8;
    #pragma unroll
    for (int t = 0; t < 3; ++t)
        #pragma unroll
        for (int v = 0; v < 8; ++v)
            outlin[(long)(m0 + moff + v) * OUT_P + t * 16 + col] = acc[t][v];
}

// ---------------------------------------------------------------- SpMM1
// out[dst,0:40] += outlin[src,0:40] * w  (16 threads/edge, 10 active float4 slices)
__global__ void gcn_spmm1(const float* __restrict__ outlin,
                          const float* __restrict__ ew,
                          const int* __restrict__ src,
                          const int* __restrict__ dst,
                          float* __restrict__ out) {
    long gid = (long)blockIdx.x * blockDim.x + threadIdx.x;
    if (gid >= (long)N_EDGES * 16) return;
    const int e = (int)(gid >> 4);
    const int r = (int)(gid & 15);
    if (r >= 10) return;
    const int f = r * 4;
    const int s = src[e], d = dst[e];
    const float w = ew[e];
    const float4 v = *(const float4*)(outlin + (long)s * OUT_P + f);
    float* o = out + (long)d * OUT_F + f;
    atomicAdd(o + 0, v.x * w);
    atomicAdd(o + 1, v.y * w);
    atomicAdd(o + 2, v.z * w);
    atomicAdd(o + 3, v.w * w);
}

// ---------------------------------------------------------------- log_softmax
// one wave32 per node over 40 logits; in-place on out, adds b1 first
__global__ __launch_bounds__(256) void gcn_lsm(float* __restrict__ out,
                                               const float* __restrict__ b1) {
    const int node = blockIdx.x * 8 + (threadIdx.x >> 5);
    if (node >= N_NODES) return;
    const int lane = threadIdx.x & 31;
    float* row = out + (long)node * OUT_F;

    float v0 = row[lane] + b1[lane];                 // lane in [0,32) < 40 always
    const bool has2 = lane < (OUT_F - 32);
    float v1 = has2 ? (row[lane + 32] + b1[lane + 32]) : -__builtin_inff();

    float m = fmaxf(v0, v1);
    #pragma unroll
    for (int off = 16; off; off >>= 1) m = fmaxf(m, __shfl_xor(m, off, 32));

    float s = expf(v0 - m) + (has2 ? expf(v1 - m) : 0.0f);
    #pragma unroll
    for (int off = 16; off; off >>= 1) s += __shfl_xor(s, off, 32);

    const float ls = m + logf(s);
    row[lane] = v0 - ls;
    if (has2) row[lane + 32] = v1 - ls;
}

// ---------------------------------------------------------------- launcher
extern "C" void kernel_launch(void* const* d_in, const int* in_sizes, int n_in,
                              void* d_out, int out_size, void* d_ws, size_t ws_size,
                              hipStream_t stream) {
    const float* x   = (const float*)d_in[0];
    const float* W0  = (const float*)d_in[1];
    const float* b0  = (const float*)d_in[2];
    const float* W1  = (const float*)d_in[3];
    const float* b1  = (const float*)d_in[4];
    const float* ew  = (const float*)d_in[5];
    const int*   src = (const int*)d_in[6];
    const int*   dst = (const int*)d_in[7];
    float* out = (float*)d_out;

    float* hlin   = (float*)d_ws;                       // [N, 128]  51.2 MB
    float* agg0   = hlin + (size_t)N_NODES * HID;       // [N, 128]  51.2 MB
    float* outlin = hlin;                               // reuse hlin region: [N, 48]

    const int gemmBlocks = (N_NODES + 127) / 128;       // 8 wave-strips of 16 rows

    // layer 0
    gcn_gemm0<<<gemmBlocks, 256, 0, stream>>>(x, W0, hlin);
    gcn_zero_f32<<<2048, 256, 0, stream>>>(agg0, (long)N_NODES * HID);
    {
        long total = (long)N_EDGES * 32;
        int blocks = (int)((total + 255) / 256);
        gcn_spmm0<<<blocks, 256, 0, stream>>>(hlin, ew, src, dst, agg0);
    }

    // layer 1 (bias+relu fused into GEMM1 A-read)
    gcn_gemm1<<<gemmBlocks, 256, 0, stream>>>(agg0, b0, W1, outlin);
    gcn_zero_f32<<<2048, 256, 0, stream>>>(out, (long)N_NODES * OUT_F);
    {
        long total = (long)N_EDGES * 16;
        int blocks = (int)((total + 255) / 256);
        gcn_spmm1<<<blocks, 256, 0, stream>>>(outlin, ew, src, dst, out);
    }

    // log_softmax (adds b1, in place)
    gcn_lsm<<<(N_NODES + 7) / 8, 256, 0, stream>>>(out, b1);
}